// ICUSTOMIntGELU_20718922236122
// MI455X (gfx1250) — compile-verified
//
#include <hip/hip_runtime.h>
#include <cmath>

// ---------------------------------------------------------------------------
// IntGELU (quantized piecewise-quadratic GELU), gfx1250 / MI455X.
// 201MB in + 201MB out, single pass -> ~17us floor at 23.3 TB/s. No matmul ->
// no WMMA (arith intensity ~0.1 flop/B). The round-2 15-compare searchsorted
// scan was VALU-bound (~37 lane-ops/elem ~ 24us > 17us mem floor); replaced
// here by an O(1) arithmetic bucketize (fma+floor+med3) with an exact +-1
// correction against division-computed bounds gathered via ds_load_b64.
// ~16 lane-ops/elem -> back under the HBM roof.
// ---------------------------------------------------------------------------

#define SEGS 16
#define TWO_N 1048576.0f  // 2^20 (N_BITS = 20)

typedef float f4 __attribute__((ext_vector_type(4)));

struct GeluConsts {
  float lo[SEGS];      // segment lower bounds (float, pre-quantization)
  float c[SEGS][3];    // polyfit coeffs, highest power first: c2, c1, c0
};

// Reference: idx = clip(searchsorted(lo_i, xi, 'right') - 1, 0, 15),
// lo_i[j] = floor(b_j / s), b_j = -5 + 0.625 j (exact fp32 values).
// Real arithmetic: idx = clamp(ceil(T)-1, 0, 15), T = (s*(xi+1)+5)/0.625.
// g = clamp(floor(fma(xi, s*1.6, (s+5)*1.6)), 0, 15) is within +-1 of idx,
// fixed by comparing against the exact lo_i neighbors from LDS.
__device__ __forceinline__ float eval_one(float xv, float inv_s,
                                          float sA, float cB,
                                          const float2* pair,  // {lo_g, lo_{g+1}}
                                          const f4* lds_c) {
  float xi = floorf(xv * inv_s);                 // floor_ste(x / s)
  float gf = floorf(fmaf(xi, sA, cB));
  gf = fminf(fmaxf(gf, 0.0f), 15.0f);            // v_med3_f32
  int g = (int)gf;
  float2 b = pair[g];                            // ds_load_b64, 8B stride
  int idx = g + ((xi >= b.y) ? 1 : 0) - ((xi < b.x) ? 1 : 0);
  f4 cc = lds_c[idx];                            // ds_load_b128, 16B stride
  float r = cc.x;                                // Horner in fp32
  r = r * xi + cc.y;
  r = r * xi + cc.z;
  return r * (1.0f / TWO_N);
}

__global__ __launch_bounds__(256) void intgelu_gfx1250_kernel(
    const float* __restrict__ x, const float* __restrict__ sp,
    float* __restrict__ out, long long n, GeluConsts K) {
  __shared__ __align__(16) float lds_c[SEGS][4];
  __shared__ __align__(8) float2 lds_pair[SEGS];

  const float s = sp[0];  // uniform -> s_load
  if (threadIdx.x == 0) {
    const float INF = __builtin_huge_valf();
    float lo[SEGS];
#pragma unroll
    for (int j = 0; j < SEGS; ++j)
      lo[j] = floorf(K.lo[j] / s);   // exact IEEE division, matches reference;
                                     // 16 independent chains pipeline
    const float s2 = s * s;
#pragma unroll
    for (int j = 0; j < SEGS; ++j) {
      // pair[g] = {lower bound of segment g, lower bound of segment g+1},
      // with -+INF sentinels so the +-1 correction self-clips at the ends.
      lds_pair[j] = make_float2(j == 0 ? -INF : lo[j],
                                j == SEGS - 1 ? INF : lo[j + 1]);
      lds_c[j][0] = floorf(K.c[j][0] * s2 * TWO_N);      // floor(c2*s^2*2^N)
      lds_c[j][1] = floorf(K.c[j][1] * s  * TWO_N);      // floor(c1*s  *2^N)
      lds_c[j][2] = floorf(K.c[j][2] * TWO_N);           // floor(c0    *2^N)
      lds_c[j][3] = 0.0f;                                // pad to 16 B
    }
  }
  __syncthreads();

  const float inv_s = 1.0f / s;                  // uniform, hoisted
  const float sA = s * 1.6f;                     // s / 0.625
  const float cB = (s + 5.0f) * 1.6f;            // (s + 5) / 0.625
  const long long n4 = n >> 2;
  const long long stride = (long long)gridDim.x * (long long)blockDim.x;
  long long i = (long long)blockIdx.x * blockDim.x + threadIdx.x;

  const f4* __restrict__ xv = (const f4*)x;
  f4* __restrict__ ov = (f4*)out;
  const f4* cc = (const f4*)lds_c;
  const float2* pr = (const float2*)lds_pair;

  for (; i < n4; i += stride) {
    // prefetch next grid-stride tile into GL2 (global_prefetch_b8); guarded
    // because WGP-scope prefetches are treated as non-speculative.
    if (i + stride < n4) __builtin_prefetch(xv + i + stride, 0, 1);
    // streaming, 402MB > 192MB L2, zero reuse -> nontemporal b128
    f4 v = __builtin_nontemporal_load(xv + i);
    f4 r;
    r.x = eval_one(v.x, inv_s, sA, cB, pr, cc);
    r.y = eval_one(v.y, inv_s, sA, cB, pr, cc);
    r.z = eval_one(v.z, inv_s, sA, cB, pr, cc);
    r.w = eval_one(v.w, inv_s, sA, cB, pr, cc);
    __builtin_nontemporal_store(r, ov + i);
  }

  if (blockIdx.x == 0 && threadIdx.x == 0) {
    for (long long e = (n4 << 2); e < n; ++e)    // tail (n % 4 != 0 safety)
      out[e] = eval_one(x[e], inv_s, sA, cB, pr, cc);
    out[n] = s * (1.0f / TWO_N);                 // scaling_factor_out = s/2^N
  }
}

// ---------------------------------------------------------------------------
// Host: least-squares degree-2 fit of GELU per segment (mirrors np.polyfit on
// 10000 samples over [-5,5], 16 segments). Deterministic; recomputed per call.
// ---------------------------------------------------------------------------
static void solve3(double M[3][3], double R[3], double out[3]) {
  for (int col = 0; col < 3; ++col) {
    int piv = col;
    for (int r = col + 1; r < 3; ++r)
      if (std::fabs(M[r][col]) > std::fabs(M[piv][col])) piv = r;
    if (piv != col) {
      for (int k = 0; k < 3; ++k) { double t = M[col][k]; M[col][k] = M[piv][k]; M[piv][k] = t; }
      double t = R[col]; R[col] = R[piv]; R[piv] = t;
    }
    for (int r = col + 1; r < 3; ++r) {
      double f = M[r][col] / M[col][col];
      for (int k = col; k < 3; ++k) M[r][k] -= f * M[col][k];
      R[r] -= f * R[col];
    }
  }
  for (int r = 2; r >= 0; --r) {
    double v = R[r];
    for (int k = r + 1; k < 3; ++k) v -= M[r][k] * out[k];
    out[r] = v / M[r][r];
  }
}

static void fit_pieces(GeluConsts* K) {
  const int NP = 10000;
  const double XLO = -5.0, XHI = 5.0;
  const double INV_SQRT2 = 0.7071067811865475244;
  for (int seg = 0; seg < SEGS; ++seg) {
    double lo = XLO + (XHI - XLO) * (double)seg / (double)SEGS;
    double hi = XLO + (XHI - XLO) * (double)(seg + 1) / (double)SEGS;
    double S0 = 0, S1 = 0, S2 = 0, S3 = 0, S4 = 0;
    double T0 = 0, T1 = 0, T2 = 0;
    for (int i = 0; i < NP; ++i) {
      double xd = XLO + (XHI - XLO) * (double)i / (double)(NP - 1);
      if (xd < lo || xd > hi) continue;
      double y = 0.5 * xd * (1.0 + std::erf(xd * INV_SQRT2));
      double x2 = xd * xd;
      S0 += 1.0; S1 += xd; S2 += x2; S3 += x2 * xd; S4 += x2 * x2;
      T0 += y; T1 += y * xd; T2 += y * x2;
    }
    double M[3][3] = {{S4, S3, S2}, {S3, S2, S1}, {S2, S1, S0}};
    double R[3] = {T2, T1, T0};
    double c[3];
    solve3(M, R, c);
    K->lo[seg] = (float)lo;
    K->c[seg][0] = (float)c[0];  // c2
    K->c[seg][1] = (float)c[1];  // c1
    K->c[seg][2] = (float)c[2];  // c0
  }
}

extern "C" void kernel_launch(void* const* d_in, const int* in_sizes, int n_in,
                              void* d_out, int out_size, void* d_ws, size_t ws_size,
                              hipStream_t stream) {
  (void)n_in; (void)out_size; (void)d_ws; (void)ws_size;
  const float* x = (const float*)d_in[0];
  const float* sp = (const float*)d_in[1];
  float* out = (float*)d_out;
  long long n = (long long)in_sizes[0];

  GeluConsts K;
  fit_pieces(&K);

  const int threads = 256;                 // 8 wave32s per block
  long long n4 = n >> 2;
  long long want = (n4 + threads - 1) / threads;
  int blocks = (int)(want < 8192 ? (want > 0 ? want : 1) : 8192);

  intgelu_gfx1250_kernel<<<blocks, threads, 0, stream>>>(x, sp, out, n, K);
}